// LatentCorrelationLayer_34368328302831
// MI455X (gfx1250) — compile-verified
//
#include <hip/hip_runtime.h>
#include <hip/hip_bf16.h>
#include <math.h>

#define EPSV 1e-8f

typedef float v2f __attribute__((ext_vector_type(2)));
typedef float v8f __attribute__((ext_vector_type(8)));

// D = A(16x4 f32) * B(4x16 f32) + C(16x16 f32), wave32 WMMA (codegen-confirmed).
__device__ __forceinline__ v8f wmma4(v2f a, v2f b, v8f c) {
  return __builtin_amdgcn_wmma_f32_16x16x4_f32(false, a, false, b, (short)0, c,
                                               false, false);
}

__device__ __forceinline__ float wave_sum32(float v) {
#pragma unroll
  for (int off = 16; off > 0; off >>= 1) v += __shfl_xor(v, off, 32);
  return v;
}

// LayerNorm over a 128-wide row (one wave per row), result into LDS dst.
__device__ __forceinline__ void ln128_row(const float* __restrict__ src,
                                          float* __restrict__ dst,
                                          const float* __restrict__ g,
                                          const float* __restrict__ b,
                                          int lane) {
  float xv[4];
#pragma unroll
  for (int j = 0; j < 4; ++j) xv[j] = src[lane + 32 * j];
  float s = xv[0] + xv[1] + xv[2] + xv[3];
  s = wave_sum32(s);
  float m = s * (1.0f / 128.0f);
  float vs = 0.0f;
#pragma unroll
  for (int j = 0; j < 4; ++j) { float d = xv[j] - m; vs += d * d; }
  vs = wave_sum32(vs);
  float inv = rsqrtf(vs * (1.0f / 128.0f) + EPSV);
#pragma unroll
  for (int j = 0; j < 4; ++j) {
    int d = lane + 32 * j;
    dst[d] = (xv[j] - m) * inv * g[d] + b[d];
  }
}

// ---- Kernel AB: fused LN1 + per-(b,d) stats + raw gram partials (WMMA) ------
// Grid: 256 x 256 (one wg per 512-row batch-eighth), dyn LDS = 128 KB.
// Gram partial for the wg's 512 rows kept in registers (v8f acc[8]/wave),
// stored with plain coalesced stores to a private 64 KB slot -> no atomics.
__global__ void k_gram(const float* __restrict__ x,
                       const float* __restrict__ g1,
                       const float* __restrict__ b1,
                       float* __restrict__ stats1,
                       float* __restrict__ stats2,
                       float* __restrict__ gpart) {
  extern __shared__ float yt[];              // [256][128] LN1'd tile
  __shared__ float s1l[128], s2l[128];
  const int tid = threadIdx.x, lane = tid & 31, wave = tid >> 5;
  const int b = blockIdx.x >> 3;             // 8 wgs per batch
  const int oct = blockIdx.x & 7;            // 512 rows each
  if (tid < 128) { s1l[tid] = 0.0f; s2l[tid] = 0.0f; }
  const int khalf = lane >> 4;
  const int nl = lane & 15;
  const v8f vzero = {0, 0, 0, 0, 0, 0, 0, 0};
  v8f acc[8];
#pragma unroll
  for (int t = 0; t < 8; ++t) acc[t] = vzero;
  float a1[4] = {0, 0, 0, 0}, a2[4] = {0, 0, 0, 0};

  for (int sc = 0; sc < 2; ++sc) {           // two 256-row sub-tiles
    const long rowbase = (long)b * 4096 + oct * 512 + sc * 256;
    __syncthreads();                         // yt reuse + s1l init visibility
    for (int i = 0; i < 32; ++i) {
      int r = wave * 32 + i;
      const float* src = x + (rowbase + r) * 128;
      float xv[4];
#pragma unroll
      for (int j = 0; j < 4; ++j) xv[j] = src[lane + 32 * j];
      float s = wave_sum32(xv[0] + xv[1] + xv[2] + xv[3]);
      float m = s * (1.0f / 128.0f);
      float vs = 0.0f;
#pragma unroll
      for (int j = 0; j < 4; ++j) { float d = xv[j] - m; vs += d * d; }
      vs = wave_sum32(vs);
      float inv = rsqrtf(vs * (1.0f / 128.0f) + EPSV);
#pragma unroll
      for (int j = 0; j < 4; ++j) {
        int d = lane + 32 * j;
        float y = (xv[j] - m) * inv * g1[d] + b1[d];
        a1[j] += y; a2[j] += y * y;
        yt[r * 128 + d] = y;
      }
    }
    __syncthreads();
    // gram accumulate: G += yt^T @ yt  (8x8 grid of 16x16 tiles, K=256)
    for (int t = 0; t < 8; ++t) {
      int idx = wave + 8 * t;                // 0..63
      int mi = idx >> 3, ni = idx & 7;
      int mrow = mi * 16 + nl;
      int ncol = ni * 16 + nl;
      v8f a = acc[t];
      for (int kb = 0; kb < 64; ++kb) {
        int k0 = kb * 4 + khalf * 2;
        v2f af, bf;
        af.x = yt[k0 * 128 + mrow];   af.y = yt[(k0 + 1) * 128 + mrow];
        bf.x = yt[k0 * 128 + ncol];   bf.y = yt[(k0 + 1) * 128 + ncol];
        a = wmma4(af, bf, a);
      }
      acc[t] = a;
    }
  }

  // store raw gram partial (no atomics)
  float* gp = gpart + (long)blockIdx.x * 16384;
  for (int t = 0; t < 8; ++t) {
    int idx = wave + 8 * t;
    int mi = idx >> 3, ni = idx & 7;
    int dmBase = mi * 16 + khalf * 8;
    int de = ni * 16 + nl;
#pragma unroll
    for (int r = 0; r < 8; ++r) gp[(dmBase + r) * 128 + de] = acc[t][r];
  }
  // stats reduction
#pragma unroll
  for (int j = 0; j < 4; ++j) {
    int d = lane + 32 * j;
    atomicAdd(&s1l[d], a1[j]);
    atomicAdd(&s2l[d], a2[j]);
  }
  __syncthreads();
  if (tid < 128) {
    atomicAdd(&stats1[b * 128 + tid], s1l[tid]);
    atomicAdd(&stats2[b * 128 + tid], s2l[tid]);
  }
}

// ---- Kernel B2: reduce gram partials + mean/norm correction -> data_corr ----
// Grid: 128 x 128 (d = block, e = thread). Streams 16 MB, no contention.
__global__ void k_combine(const float* __restrict__ stats1,
                          const float* __restrict__ stats2,
                          const float* __restrict__ gpart,
                          float* __restrict__ dcorr) {
  const int d = blockIdx.x;
  const int e = threadIdx.x;
  float acc = 0.0f;
  for (int b = 0; b < 32; ++b) {
    float s1d = stats1[b * 128 + d], s2d = stats2[b * 128 + d];
    float s1e = stats1[b * 128 + e], s2e = stats2[b * 128 + e];
    float md = s1d * (1.0f / 4096.0f), me = s1e * (1.0f / 4096.0f);
    float vd = s2d - s1d * s1d * (1.0f / 4096.0f);
    float ve = s2e - s1e * s1e * (1.0f / 4096.0f);
    float ind = 1.0f / fmaxf(sqrtf(fmaxf(vd, 0.0f)), EPSV);
    float ine = 1.0f / fmaxf(sqrtf(fmaxf(ve, 0.0f)), EPSV);
    float G = 0.0f;
#pragma unroll
    for (int q = 0; q < 8; ++q)
      G += gpart[(long)(b * 8 + q) * 16384 + d * 128 + e];
    acc += (G - 4096.0f * md * me) * ind * ine;
  }
  dcorr[d * 128 + e] = acc * (1.0f / 32.0f);
}

// --------- Kernel C: mixed correlation, Laplacian, Chebyshev softmax ---------
__global__ void k_graph(const float* __restrict__ corr,
                        const float* __restrict__ alpha_logit,
                        const float* __restrict__ chebw,
                        const float* __restrict__ dcorr,
                        float* __restrict__ mixed_out,
                        float* __restrict__ Lap,
                        float* __restrict__ wout) {
  __shared__ float dis[128];
  const int i = threadIdx.x;                 // 0..127, one row each
  const float alpha = 1.0f / (1.0f + __expf(-alpha_logit[0]));
  float deg = 0.0f;
  for (int j = 0; j < 128; ++j) {
    float learned = tanhf(0.5f * (corr[i * 128 + j] + corr[j * 128 + i]));
    float dc = fminf(fmaxf(dcorr[i * 128 + j], -1.0f), 1.0f);
    float mx = alpha * learned + (1.0f - alpha) * dc;
    mixed_out[i * 128 + j] = mx;
    if (j != i) deg += mx;
  }
  dis[i] = rsqrtf(fmaxf(deg, EPSV));
  __syncthreads();
  const float di = dis[i];
  for (int j = 0; j < 128; ++j) {
    float a = (j == i) ? 0.0f : mixed_out[i * 128 + j];
    float l = ((j == i) ? 1.0f : 0.0f) - di * a * dis[j];
    Lap[i * 128 + j] = fminf(fmaxf(l, -1.5f), 1.5f);
  }
  if (i == 0) {
    float c0 = chebw[0], c1 = chebw[1], c2 = chebw[2];
    float mx = fmaxf(c0, fmaxf(c1, c2));
    float e0 = __expf(c0 - mx), e1 = __expf(c1 - mx), e2 = __expf(c2 - mx);
    float s = e0 + e1 + e2;
    wout[0] = e0 / s; wout[1] = e1 / s; wout[2] = e2 / s;
  }
}

// -------- Kernel D: fused LN1 -> Chebyshev -> Wi/LN2/GELU -> Wo/LN3 ----------
// Grid: 2048 x 256, dyn LDS = 128 KB (yo 32K | t1 32K | h 64K).
__global__ void k_main(const float* __restrict__ x,
                       const float* __restrict__ g1, const float* __restrict__ b1,
                       const float* __restrict__ Lap, const float* __restrict__ wv,
                       const float* __restrict__ Wi, const float* __restrict__ bi,
                       const float* __restrict__ Wo, const float* __restrict__ bo,
                       const float* __restrict__ g2, const float* __restrict__ b2,
                       const float* __restrict__ g3, const float* __restrict__ b3,
                       float* __restrict__ out) {
  extern __shared__ float smem[];
  float* yo = smem;                // [64][128] : y, later Chebyshev output
  float* t1 = smem + 8192;         // [64][128] : Tx1, later o pre-LN
  float* hb = smem + 16384;        // [64][256] : hidden
  const int tid = threadIdx.x, lane = tid & 31, wave = tid >> 5;
  const long base = (long)blockIdx.x * 64;
  const float w0 = wv[0], w1 = wv[1], w2 = wv[2];
  const int khalf = lane >> 4;
  const int nl = lane & 15;

  // LN1
  for (int i = 0; i < 8; ++i) {
    int r = wave * 8 + i;
    ln128_row(x + (base + r) * 128, yo + r * 128, g1, b1, lane);
  }
  __syncthreads();

  // Tx1 = yo @ Lap   (4x8 tiles, K=128)
  for (int t = 0; t < 4; ++t) {
    int idx = wave + 8 * t; int mi = idx >> 3, ni = idx & 7;
    v8f acc = {};
    int mrow = mi * 16 + nl, nc = ni * 16 + nl;
    for (int kb = 0; kb < 32; ++kb) {
      int k0 = kb * 4 + khalf * 2;
      v2f a, bb;
      a.x = yo[mrow * 128 + k0];  a.y = yo[mrow * 128 + k0 + 1];
      bb.x = Lap[k0 * 128 + nc];  bb.y = Lap[(k0 + 1) * 128 + nc];
      acc = wmma4(a, bb, acc);
    }
    int rowB = mi * 16 + khalf * 8;
#pragma unroll
    for (int r = 0; r < 8; ++r) t1[(rowB + r) * 128 + nc] = acc[r];
  }
  __syncthreads();

  // Tx2 = clip(2*Tx1@Lap - y); out = w0*y + w1*Tx1 + w2*Tx2  (into yo)
  for (int t = 0; t < 4; ++t) {
    int idx = wave + 8 * t; int mi = idx >> 3, ni = idx & 7;
    v8f acc = {};
    int mrow = mi * 16 + nl, nc = ni * 16 + nl;
    for (int kb = 0; kb < 32; ++kb) {
      int k0 = kb * 4 + khalf * 2;
      v2f a, bb;
      a.x = t1[mrow * 128 + k0];  a.y = t1[mrow * 128 + k0 + 1];
      bb.x = Lap[k0 * 128 + nc];  bb.y = Lap[(k0 + 1) * 128 + nc];
      acc = wmma4(a, bb, acc);
    }
    int rowB = mi * 16 + khalf * 8;
#pragma unroll
    for (int r = 0; r < 8; ++r) {
      int pos = (rowB + r) * 128 + nc;
      float y0 = yo[pos], y1 = t1[pos];
      float tx2 = fminf(fmaxf(2.0f * acc[r] - y0, -50.0f), 50.0f);
      yo[pos] = w0 * y0 + w1 * y1 + w2 * tx2;   // lane-exclusive position
    }
  }
  __syncthreads();

  // h = out @ Wi^T + bi   (4x16 tiles, K=128)
  for (int t = 0; t < 8; ++t) {
    int idx = wave + 8 * t; int mi = idx >> 4, ni = idx & 15;
    v8f acc = {};
    int mrow = mi * 16 + nl, nc = ni * 16 + nl;
    for (int kb = 0; kb < 32; ++kb) {
      int k0 = kb * 4 + khalf * 2;
      v2f a, bb;
      a.x = yo[mrow * 128 + k0];  a.y = yo[mrow * 128 + k0 + 1];
      bb.x = Wi[nc * 128 + k0];   bb.y = Wi[nc * 128 + k0 + 1];
      acc = wmma4(a, bb, acc);
    }
    int rowB = mi * 16 + khalf * 8;
    float bia = bi[nc];
#pragma unroll
    for (int r = 0; r < 8; ++r) hb[(rowB + r) * 256 + nc] = acc[r] + bia;
  }
  __syncthreads();

  // LN2 + exact GELU (rows of 256)
  for (int i = 0; i < 8; ++i) {
    int r = wave * 8 + i;
    float xv[8];
#pragma unroll
    for (int j = 0; j < 8; ++j) xv[j] = hb[r * 256 + lane + 32 * j];
    float s = 0.0f;
#pragma unroll
    for (int j = 0; j < 8; ++j) s += xv[j];
    s = wave_sum32(s);
    float m = s * (1.0f / 256.0f);
    float vs = 0.0f;
#pragma unroll
    for (int j = 0; j < 8; ++j) { float d = xv[j] - m; vs += d * d; }
    vs = wave_sum32(vs);
    float inv = rsqrtf(vs * (1.0f / 256.0f) + EPSV);
#pragma unroll
    for (int j = 0; j < 8; ++j) {
      int d = lane + 32 * j;
      float yv = (xv[j] - m) * inv * g2[d] + b2[d];
      hb[r * 256 + d] = 0.5f * yv * (1.0f + erff(yv * 0.70710678118654752f));
    }
  }
  __syncthreads();

  // o = h @ Wo^T + bo   (4x8 tiles, K=256) -> t1
  for (int t = 0; t < 4; ++t) {
    int idx = wave + 8 * t; int mi = idx >> 3, ni = idx & 7;
    v8f acc = {};
    int mrow = mi * 16 + nl, nc = ni * 16 + nl;
    for (int kb = 0; kb < 64; ++kb) {
      int k0 = kb * 4 + khalf * 2;
      v2f a, bb;
      a.x = hb[mrow * 256 + k0];  a.y = hb[mrow * 256 + k0 + 1];
      bb.x = Wo[nc * 256 + k0];   bb.y = Wo[nc * 256 + k0 + 1];
      acc = wmma4(a, bb, acc);
    }
    int rowB = mi * 16 + khalf * 8;
    float boa = bo[nc];
#pragma unroll
    for (int r = 0; r < 8; ++r) t1[(rowB + r) * 128 + nc] = acc[r] + boa;
  }
  __syncthreads();

  // LN3, store to global
  for (int i = 0; i < 8; ++i) {
    int r = wave * 8 + i;
    float xv[4];
#pragma unroll
    for (int j = 0; j < 4; ++j) xv[j] = t1[r * 128 + lane + 32 * j];
    float s = wave_sum32(xv[0] + xv[1] + xv[2] + xv[3]);
    float m = s * (1.0f / 128.0f);
    float vs = 0.0f;
#pragma unroll
    for (int j = 0; j < 4; ++j) { float d = xv[j] - m; vs += d * d; }
    vs = wave_sum32(vs);
    float inv = rsqrtf(vs * (1.0f / 128.0f) + EPSV);
#pragma unroll
    for (int j = 0; j < 4; ++j) {
      int d = lane + 32 * j;
      out[(base + r) * 128 + d] = (xv[j] - m) * inv * g3[d] + b3[d];
    }
  }
}

extern "C" void kernel_launch(void* const* d_in, const int* in_sizes, int n_in,
                              void* d_out, int out_size, void* d_ws, size_t ws_size,
                              hipStream_t stream) {
  const float* x     = (const float*)d_in[0];
  const float* corr  = (const float*)d_in[1];
  const float* alog  = (const float*)d_in[2];
  const float* chebw = (const float*)d_in[3];
  const float* Wi    = (const float*)d_in[4];
  const float* bi    = (const float*)d_in[5];
  const float* Wo    = (const float*)d_in[6];
  const float* bo    = (const float*)d_in[7];
  const float* g1    = (const float*)d_in[8];
  const float* b1    = (const float*)d_in[9];
  const float* g2    = (const float*)d_in[10];
  const float* b2    = (const float*)d_in[11];
  const float* g3    = (const float*)d_in[12];
  const float* b3    = (const float*)d_in[13];
  float* out = (float*)d_out;
  float* ws  = (float*)d_ws;

  float* stats1 = ws;                       // [32*128]
  float* stats2 = ws + 4096;                // [32*128]
  float* gpart  = ws + 8192;                // [256][128*128] raw gram partials
  float* dcorr  = ws + 8192 + 256 * 16384;  // [128*128]
  float* Lap    = dcorr + 16384;            // [128*128]
  float* wv     = Lap + 16384;              // [3]
  float* mixed  = out + (long)32 * 4096 * 128;  // second tuple element

  hipMemsetAsync(d_ws, 0, 8192 * sizeof(float), stream);  // stats only
  k_gram<<<256, 256, 256 * 128 * sizeof(float), stream>>>(x, g1, b1, stats1,
                                                          stats2, gpart);
  k_combine<<<128, 128, 0, stream>>>(stats1, stats2, gpart, dcorr);
  k_graph<<<1, 128, 0, stream>>>(corr, alog, chebw, dcorr, mixed, Lap, wv);
  k_main<<<2048, 256, 131072, stream>>>(x, g1, b1, Lap, wv, Wi, bi, Wo, bo,
                                        g2, b2, g3, b3, out);
}